// DilateLoss_66005057405719
// MI455X (gfx1250) — compile-verified
//
#include <hip/hip_runtime.h>
#include <math.h>

// ---------------- problem constants (match reference) ----------------
#define NB      64          // batches
#define NT      24          // time steps (DTW N)
#define HW      65536       // 256*256
#define PIXELS  50
#define BSEQ    (NB*PIXELS) // 3200 sequences
#define ALPHA   0.1f
#define GAMMA   0.001f
#define INVG    (1.0f/GAMMA)
#define BIGV    1.0e8f
#define NEGPAD  (-1.0e30f)  // backward-pass sentinel: exp((NEGPAD-x)/g) == 0

// ---------------- workspace layout (bytes) ----------------
#define WS_TOPIDX 0          // int   topidx[3200]
#define WS_SHAPE  16384      // float pshape[400]
#define WS_PATH   20480      // float ppath[400][576]
#define NBLK_B    (BSEQ/8)   // 400 blocks, 8 waves (sequences) each

// =====================================================================
// Kernel A: exact top-50 per batch (tournament selection, deterministic)
// =====================================================================
__global__ __launch_bounds__(1024)
void topk_kernel(const float* __restrict__ sc, int* __restrict__ topidx)
{
    __shared__ float cV[1024];
    __shared__ int   cI[1024];
    __shared__ int   cT[1024];
    const int b   = blockIdx.x;
    const int tid = threadIdx.x;
    const float* base = sc + (size_t)b * HW;

    // "last selected" key for THIS thread's chunk; init = +inf sentinel
    float lastV = 3.402823466e38f; int lastI = -1;

    // initial candidate: best element in my strided 64-element chunk
    float bv = -3.402823466e38f; int bi = 0x7fffffff;
    for (int e = 0; e < HW/1024; ++e) {
        int idx = tid + e*1024;
        float v = base[idx];
        bool worse = (v < lastV) || (v == lastV && idx > lastI);
        if (worse && ((v > bv) || (v == bv && idx < bi))) { bv = v; bi = idx; }
    }

    for (int r = 0; r < PIXELS; ++r) {
        cV[tid] = bv; cI[tid] = bi; cT[tid] = tid;
        __syncthreads();
        // deterministic argmax tree over 1024 candidates, key = (v desc, idx asc)
        for (int s = 512; s > 0; s >>= 1) {
            if (tid < s) {
                float ov = cV[tid+s]; int oi = cI[tid+s];
                if ((ov > cV[tid]) || (ov == cV[tid] && oi < cI[tid])) {
                    cV[tid] = ov; cI[tid] = oi; cT[tid] = cT[tid+s];
                }
            }
            __syncthreads();
        }
        float selV = cV[0]; int selI = cI[0]; int winner = cT[0];
        if (tid == 0) topidx[b*PIXELS + r] = selI;
        __syncthreads();                 // everyone has read slot 0
        if (tid == winner) {             // winner advances & rescans its chunk
            lastV = selV; lastI = selI;
            bv = -3.402823466e38f; bi = 0x7fffffff;
            for (int e = 0; e < HW/1024; ++e) {
                int idx = tid + e*1024;
                float v = base[idx];     // L2-resident after first pass
                bool worse = (v < lastV) || (v == lastV && idx > lastI);
                if (worse && ((v > bv) || (v == bv && idx < bi))) { bv=v; bi=idx; }
            }
        }
    }
}

// =====================================================================
// Kernel B: soft-DTW forward + exact adjoint, one wave32 per sequence
// =====================================================================
#define WV   8                 // waves (sequences) per block
#define RSZ  (26*26)           // R / E padded to 26x26
#define WSTR (RSZ*2 + 56)      // per-wave LDS floats: R, E, t[25], p[25] (+pad)

__global__ __launch_bounds__(256)
void dtw_kernel(const float* __restrict__ pred,
                const float* __restrict__ truev,
                const int*   __restrict__ topidx,
                float* __restrict__ pshape,
                float* __restrict__ ppath)
{
    __shared__ float sm[WV*WSTR];
    __shared__ float wshape[WV];
    const int tid  = threadIdx.x;
    const int w    = tid >> 5;
    const int lane = tid & 31;
    float* R  = sm + w*WSTR;
    float* E  = R + RSZ;
    float* tv = E + RSZ;       // 25 entries (index 24 = pad 0)
    float* pv = tv + 25;       // 25 entries

    const int k   = blockIdx.x*WV + w;          // 3200 = 400*8 exactly
    const int b   = k / PIXELS;
    const int pix = topidx[k];

    // init: R = BIG everywhere, R[0][0]=0, row/col 25 = -1e30 (backward pad), E = 0
    for (int c = lane; c < RSZ; c += 32) { R[c] = BIGV; E[c] = 0.0f; }
    if (lane == 0) R[0] = 0.0f;
    if (lane < 26) { R[25*26 + lane] = NEGPAD; R[lane*26 + 25] = NEGPAD; }

    // gather the 24-step sequences (scattered, tiny volume)
    if (lane < NT) {
        size_t off = ((size_t)(b*NT + lane))*HW + (size_t)pix;
        __builtin_prefetch(&truev[off], 0, 1);   // global_prefetch_b8
        __builtin_prefetch(&pred[off],  0, 1);
        tv[lane] = truev[off];
        pv[lane] = pred[off];
    }
    if (lane == 0) { tv[24] = 0.0f; pv[24] = 0.0f; }
    __syncthreads();

    const int i = lane + 1;     // 1-based row handled by this lane

    // ---- forward wavefront: R[i][j] = D + softmin(diag, up, left) ----
    for (int d = 2; d <= 2*NT; ++d) {
        int j = d - i;
        if (lane < NT && j >= 1 && j <= NT) {
            float diag = R[(i-1)*26 + (j-1)];
            float up   = R[(i-1)*26 + j];
            float left = R[i*26 + (j-1)];
            float mn   = fminf(diag, fminf(up, left));
            float ssum = __expf(-(diag-mn)*INVG) + __expf(-(up-mn)*INVG)
                       + __expf(-(left-mn)*INVG);
            float dt   = tv[i-1] - pv[j-1];
            R[i*26 + j] = dt*dt + mn - GAMMA*__logf(ssum);
        }
        __syncthreads();
    }

    // ---- backward wavefront: exact adjoint E (== jax.grad of the scan) ----
    for (int d = 2*NT; d >= 2; --d) {
        int j = d - i;
        if (lane < NT && j >= 1 && j <= NT) {
            float ev;
            if (i == NT && j == NT) {
                ev = 1.0f;
            } else {
                float Rv  = R[i*26 + j];
                float d10 = tv[i]   - pv[j-1];   // D(i+1, j)
                float d01 = tv[i-1] - pv[j];     // D(i, j+1)
                float d11 = tv[i]   - pv[j];     // D(i+1, j+1)
                float wa = __expf((R[(i+1)*26 + j]     - Rv - d10*d10)*INVG);
                float wb = __expf((R[i*26 + (j+1)]     - Rv - d01*d01)*INVG);
                float wc = __expf((R[(i+1)*26 + (j+1)] - Rv - d11*d11)*INVG);
                ev = wa*E[(i+1)*26 + j] + wb*E[i*26 + (j+1)] + wc*E[(i+1)*26 + (j+1)];
            }
            E[i*26 + j] = ev;
        }
        __syncthreads();
    }

    // ---- deterministic per-block partials ----
    if (lane == 0) wshape[w] = R[NT*26 + NT];
    __syncthreads();
    if (tid == 0) {
        float s = 0.0f;
        for (int q = 0; q < WV; ++q) s += wshape[q];
        pshape[blockIdx.x] = s;
    }
    for (int c = tid; c < NT*NT; c += 256) {
        int ii = c / NT, jj = c % NT;
        float s = 0.0f;
        for (int q = 0; q < WV; ++q)
            s += sm[q*WSTR + RSZ + (ii+1)*26 + (jj+1)];
        ppath[(size_t)blockIdx.x*(NT*NT) + c] = s;
    }
}

// =====================================================================
// Kernel C: deterministic final reduce + WMMA contraction
//   sum(path * Omega) = trace(P * Omega^T), Omega[i][j] = (i-j)^2 (symmetric)
//   Done with V_WMMA_F32_16X16X4_F32 on P zero-padded to 32x32.
// =====================================================================
typedef __attribute__((ext_vector_type(2))) float v2f;
typedef __attribute__((ext_vector_type(8))) float v8f;

__global__ __launch_bounds__(256)
void finalize_kernel(const float* __restrict__ pshape,
                     const float* __restrict__ ppath,
                     float* __restrict__ out)
{
    __shared__ float P[32*32];      // path sum, zero padded
    __shared__ float CT[2*16*16];   // two diagonal C tiles
    __shared__ float shapeSum;
    const int tid = threadIdx.x;

    for (int c = tid; c < 32*32; c += 256) P[c] = 0.0f;
    __syncthreads();

    for (int c = tid; c < NT*NT; c += 256) {       // fixed-order block sum
        int ii = c / NT, jj = c % NT;
        float s = 0.0f;
        for (int blk = 0; blk < NBLK_B; ++blk)
            s += ppath[(size_t)blk*(NT*NT) + c];
        P[ii*32 + jj] = s;
    }
    if (tid == 0) {
        float s = 0.0f;
        for (int blk = 0; blk < NBLK_B; ++blk) s += pshape[blk];
        shapeSum = s;
    }
    __syncthreads();

    if (tid < 32) {                                 // wave 0: EXEC all-ones
        const int lane = tid;
        const int half = lane >> 4;                 // A/B lane-half per ISA layout
        const int lmod = lane & 15;
        for (int T = 0; T < 2; ++T) {               // diagonal tiles only
            v8f acc = {};
            const int mrow = lmod + 16*T;           // A row == B col (diag tile)
            for (int kc = 0; kc < 6; ++kc) {        // K = 24 in chunks of 4
                int kb = kc*4;
                // A 16x4 f32 layout: lanes0-15 VGPR{0,1}=K{0,1}; lanes16-31 K{2,3}
                int ka0 = kb + (half ? 2 : 0);
                v2f a; a.x = P[mrow*32 + ka0]; a.y = P[mrow*32 + ka0 + 1];
                // B 4x16 f32 layout: VGPR0 rows {K0|K2}, VGPR1 rows {K1|K3}
                int kb0 = kb + (half ? 2 : 0);
                int kb1 = kb + (half ? 3 : 1);
                float dn0 = (float)(mrow - kb0);
                float dn1 = (float)(mrow - kb1);
                v2f bf; bf.x = dn0*dn0; bf.y = dn1*dn1;   // Omega (symmetric)
                acc = __builtin_amdgcn_wmma_f32_16x16x4_f32(
                          false, a, false, bf, (short)0, acc, false, false);
            }
            // C layout: VGPR v -> M = v (+8 for lanes16-31), N = lane&15
            for (int v = 0; v < 8; ++v) {
                int Ml = v + (half ? 8 : 0);
                CT[T*256 + Ml*16 + lmod] = acc[v];
            }
        }
    }
    __syncthreads();

    if (tid == 0) {
        float tr = 0.0f;                            // trace(P * Omega^T)
        for (int d = 0; d < 16; ++d)
            tr += CT[d*16 + d] + CT[256 + d*16 + d];
        float loss = ALPHA * (shapeSum / (float)BSEQ)
                   + (1.0f - ALPHA) * (tr / ((float)BSEQ * (float)(NT*NT)));
        out[0] = loss;
    }
}

// =====================================================================
extern "C" void kernel_launch(void* const* d_in, const int* in_sizes, int n_in,
                              void* d_out, int out_size, void* d_ws, size_t ws_size,
                              hipStream_t stream)
{
    (void)in_sizes; (void)n_in; (void)out_size; (void)ws_size;
    const float* pred  = (const float*)d_in[0];
    const float* truev = (const float*)d_in[1];
    const float* sc    = (const float*)d_in[2];

    int*   topidx = (int*)  ((char*)d_ws + WS_TOPIDX);
    float* pshape = (float*)((char*)d_ws + WS_SHAPE);
    float* ppath  = (float*)((char*)d_ws + WS_PATH);

    topk_kernel    <<<NB,     1024, 0, stream>>>(sc, topidx);
    dtw_kernel     <<<NBLK_B,  256, 0, stream>>>(pred, truev, topidx, pshape, ppath);
    finalize_kernel<<<1,       256, 0, stream>>>(pshape, ppath, (float*)d_out);
}